// GAT_64252710748438
// MI455X (gfx1250) — compile-verified
//
#include <hip/hip_runtime.h>
#include <hip/hip_bf16.h>
#include <math.h>

typedef float v2f __attribute__((ext_vector_type(2)));
typedef float v8f __attribute__((ext_vector_type(8)));

namespace {
constexpr int kN    = 50000;
constexpr int kE    = 800000;
constexpr int kEtot = kE + kN;     // self-loops appended
constexpr int kInCh = 128;
constexpr int kD1   = 128;         // HEADS*HID
constexpr int kHeads = 4;
constexpr int kHid  = 32;
constexpr int kOut  = 40;
constexpr int kOutPad = 48;        // 3 x 16-col tiles, zero padded
constexpr int kLdsStride = 132;    // 128 + 4 pad -> conflict-free A reads
constexpr float kLeaky = 0.2f;
constexpr float kBnEps = 1e-5f;
}

// ---- monotone float<->uint encoding for atomicMax on floats ----------------
__device__ __forceinline__ unsigned ord_enc(float f) {
  unsigned u = __float_as_uint(f);
  return (u & 0x80000000u) ? ~u : (u | 0x80000000u);
}
__device__ __forceinline__ float ord_dec(unsigned u) {
  return (u & 0x80000000u) ? __uint_as_float(u & 0x7fffffffu)
                           : __uint_as_float(~u);
}

__device__ __forceinline__ void edge_sd(const int* __restrict__ ei, int i,
                                        int& s, int& d) {
  if (i < kE) { s = ei[i]; d = ei[kE + i]; }
  else        { s = i - kE; d = i - kE; }     // self-loop
}

// ---- workspace zero / -inf init (deterministic each call) ------------------
__global__ void init_ws_kernel(float* agg1, float* agg2, float* sum1,
                               float* sum2, unsigned* mx1, unsigned* mx2) {
  int i = blockIdx.x * blockDim.x + threadIdx.x;
  if (i < kN * kD1)    agg1[i] = 0.f;
  if (i < kN * kOut)   agg2[i] = 0.f;
  if (i < kN * kHeads) { sum1[i] = 0.f; mx1[i] = 0u; }
  if (i < kN)          { sum2[i] = 0.f; mx2[i] = 0u; }
}

// ---- weight transposes (once per call; makes B-operand pairs contiguous) ---
__global__ void transpose_w1(const float* __restrict__ W, float* __restrict__ Wt) {
  int i = blockIdx.x * blockDim.x + threadIdx.x;     // dest index c*128+k
  if (i >= kD1 * kInCh) return;
  int c = i >> 7, k = i & 127;
  Wt[i] = W[k * kD1 + c];
}
__global__ void transpose_w2(const float* __restrict__ W2, float* __restrict__ W2t) {
  int i = blockIdx.x * blockDim.x + threadIdx.x;     // dest index c*128+k
  if (i >= kOutPad * kD1) return;
  int c = i >> 7, k = i & 127;
  W2t[i] = (c < kOut) ? W2[k * kOut + c] : 0.f;
}

// ---- layer-1 GEMM: h1 = x @ W1  (f32 WMMA 16x16x4) -------------------------
// block = 128 threads (4 waves). A row-block staged in LDS (shared by all
// waves); each wave owns two 16-col tiles and reuses A across both.
__global__ void gemm1_wmma(const float* __restrict__ x,
                           const float* __restrict__ Wt,   // [128 col][128 k]
                           float* __restrict__ h1) {
  __shared__ float lds_a[16 * kLdsStride];
  const int tid = threadIdx.x;
  const int rowBase = blockIdx.x * 16;
  for (int i = tid; i < 16 * kInCh; i += 128) {
    int r = i >> 7, k = i & 127;
    lds_a[r * kLdsStride + k] = x[(rowBase + r) * kInCh + k];
  }
  __syncthreads();

  const int lane = tid & 31, wave = tid >> 5;
  const int halfSel = lane >> 4;            // 0 = lanes 0-15, 1 = lanes 16-31
  const int lr = lane & 15;
  const int col0 = wave * 32 + lr;          // tile 0 column
  const int col1 = col0 + 16;               // tile 1 column
  v8f acc0 = {0.f, 0.f, 0.f, 0.f, 0.f, 0.f, 0.f, 0.f};
  v8f acc1 = {0.f, 0.f, 0.f, 0.f, 0.f, 0.f, 0.f, 0.f};
  for (int k0 = 0; k0 < kInCh; k0 += 4) {
    const int kb = k0 + halfSel * 2;
    v2f a  = *reinterpret_cast<const v2f*>(&lds_a[lr * kLdsStride + kb]); // ds b64
    v2f b0 = *reinterpret_cast<const v2f*>(&Wt[col0 * kInCh + kb]);       // glb b64
    v2f b1 = *reinterpret_cast<const v2f*>(&Wt[col1 * kInCh + kb]);
    acc0 = __builtin_amdgcn_wmma_f32_16x16x4_f32(false, a, false, b0,
                                                 (short)0, acc0, false, false);
    acc1 = __builtin_amdgcn_wmma_f32_16x16x4_f32(false, a, false, b1,
                                                 (short)0, acc1, false, false);
  }
#pragma unroll
  for (int r = 0; r < 8; ++r) {
    const int mrow = rowBase + r + halfSel * 8;
    h1[mrow * kD1 + col0] = acc0[r];
    h1[mrow * kD1 + col1] = acc1[r];
  }
}

// ---- layer-2 GEMM: h2 = h @ W2  (1 wave/block, 3 col tiles, A in LDS) ------
__global__ void gemm2_wmma(const float* __restrict__ h,
                           const float* __restrict__ W2t,  // [48 col][128 k]
                           float* __restrict__ h2) {
  __shared__ float lds_a[16 * kLdsStride];
  const int lane = threadIdx.x;             // 32 threads
  const int rowBase = blockIdx.x * 16;
  for (int i = lane; i < 16 * kD1; i += 32) {
    int r = i >> 7, k = i & 127;
    lds_a[r * kLdsStride + k] = h[(rowBase + r) * kD1 + k];
  }
  __syncthreads();

  const int halfSel = lane >> 4;
  const int lr = lane & 15;
  v8f acc[3];
#pragma unroll
  for (int t = 0; t < 3; ++t)
    acc[t] = (v8f){0.f, 0.f, 0.f, 0.f, 0.f, 0.f, 0.f, 0.f};
  for (int k0 = 0; k0 < kD1; k0 += 4) {
    const int kb = k0 + halfSel * 2;
    v2f a = *reinterpret_cast<const v2f*>(&lds_a[lr * kLdsStride + kb]);
#pragma unroll
    for (int t = 0; t < 3; ++t) {
      v2f b = *reinterpret_cast<const v2f*>(&W2t[(t * 16 + lr) * kD1 + kb]);
      acc[t] = __builtin_amdgcn_wmma_f32_16x16x4_f32(false, a, false, b,
                                                     (short)0, acc[t], false, false);
    }
  }
#pragma unroll
  for (int t = 0; t < 3; ++t) {
    const int col = t * 16 + lr;
    if (col < kOut) {
#pragma unroll
      for (int r = 0; r < 8; ++r)
        h2[(rowBase + r + halfSel * 8) * kOut + col] = acc[t][r];
    }
  }
}

// ---- per-node attention scores ---------------------------------------------
__global__ void scores1_kernel(const float* __restrict__ h1,
                               const float* __restrict__ a_src,
                               const float* __restrict__ a_dst,
                               float* __restrict__ es, float* __restrict__ ed) {
  int idx = blockIdx.x * blockDim.x + threadIdx.x;   // n*HEADS + h
  if (idx >= kN * kHeads) return;
  int n = idx >> 2, hh = idx & 3;
  const float* hp = h1 + n * kD1 + hh * kHid;
  const float* as = a_src + hh * kHid;
  const float* ad = a_dst + hh * kHid;
  float ss = 0.f, sd = 0.f;
#pragma unroll
  for (int c = 0; c < kHid; ++c) { float v = hp[c]; ss += v * as[c]; sd += v * ad[c]; }
  es[idx] = ss; ed[idx] = sd;
}

__global__ void scores2_kernel(const float* __restrict__ h2,
                               const float* __restrict__ a_src,
                               const float* __restrict__ a_dst,
                               float* __restrict__ es, float* __restrict__ ed) {
  int n = blockIdx.x * blockDim.x + threadIdx.x;
  if (n >= kN) return;
  const float* hp = h2 + n * kOut;
  float ss = 0.f, sd = 0.f;
#pragma unroll
  for (int c = 0; c < kOut; ++c) { float v = hp[c]; ss += v * a_src[c]; sd += v * a_dst[c]; }
  es[n] = ss; ed[n] = sd;
}

// ---- segment-softmax passes, layer 1 ---------------------------------------
__global__ void edge_max1(const int* __restrict__ ei, const float* __restrict__ es,
                          const float* __restrict__ ed, unsigned* __restrict__ mx) {
  int i = blockIdx.x * blockDim.x + threadIdx.x;
  if (i >= kEtot) return;
  int s, d; edge_sd(ei, i, s, d);
#pragma unroll
  for (int h = 0; h < kHeads; ++h) {
    float e = es[s * kHeads + h] + ed[d * kHeads + h];
    e = e > 0.f ? e : kLeaky * e;
    atomicMax(&mx[d * kHeads + h], ord_enc(e));
  }
}

__global__ void edge_sum1(const int* __restrict__ ei, const float* __restrict__ es,
                          const float* __restrict__ ed, const unsigned* __restrict__ mx,
                          float* __restrict__ sum) {
  int i = blockIdx.x * blockDim.x + threadIdx.x;
  if (i >= kEtot) return;
  int s, d; edge_sd(ei, i, s, d);
#pragma unroll
  for (int h = 0; h < kHeads; ++h) {
    float e = es[s * kHeads + h] + ed[d * kHeads + h];
    e = e > 0.f ? e : kLeaky * e;
    float ex = expf(e - ord_dec(mx[d * kHeads + h]));
    atomicAdd(&sum[d * kHeads + h], ex);
  }
}

// 32 lanes per edge; lane c handles features [4c,4c+4) -> head = c>>3
__global__ void edge_agg1(const int* __restrict__ ei, const float* __restrict__ es,
                          const float* __restrict__ ed, const unsigned* __restrict__ mx,
                          const float* __restrict__ h1, float* __restrict__ agg) {
  int lane = threadIdx.x & 31;
  int i = blockIdx.x * (blockDim.x >> 5) + (threadIdx.x >> 5);
  if (i >= kEtot) return;
  int s, d; edge_sd(ei, i, s, d);
  int c0 = lane * 4;
  int hh = c0 >> 5;
  float e = es[s * kHeads + hh] + ed[d * kHeads + hh];
  e = e > 0.f ? e : kLeaky * e;
  float ex = expf(e - ord_dec(mx[d * kHeads + hh]));
  const float4 hv = *reinterpret_cast<const float4*>(h1 + s * kD1 + c0);
  float* ap = agg + d * kD1 + c0;
  atomicAdd(ap + 0, hv.x * ex);
  atomicAdd(ap + 1, hv.y * ex);
  atomicAdd(ap + 2, hv.z * ex);
  atomicAdd(ap + 3, hv.w * ex);
}

// normalize + bias + BN(eval) + ELU, write layer-2 input (reuses h1 buffer)
__global__ void finalize1(const float* __restrict__ agg, const float* __restrict__ sum,
                          const float* __restrict__ b1, const float* __restrict__ g,
                          const float* __restrict__ be, const float* __restrict__ mu,
                          const float* __restrict__ var, float* __restrict__ out) {
  int idx = blockIdx.x * blockDim.x + threadIdx.x;
  if (idx >= kN * kD1) return;
  int n = idx / kD1, j = idx - n * kD1;
  int hh = j >> 5;
  float v = agg[idx] / sum[n * kHeads + hh] + b1[j];
  v = (v - mu[j]) * rsqrtf(var[j] + kBnEps) * g[j] + be[j];
  v = v > 0.f ? v : expf(v) - 1.f;                    // ELU
  out[idx] = v;
}

// ---- segment-softmax passes, layer 2 (1 head) ------------------------------
__global__ void edge_max2(const int* __restrict__ ei, const float* __restrict__ es,
                          const float* __restrict__ ed, unsigned* __restrict__ mx) {
  int i = blockIdx.x * blockDim.x + threadIdx.x;
  if (i >= kEtot) return;
  int s, d; edge_sd(ei, i, s, d);
  float e = es[s] + ed[d];
  e = e > 0.f ? e : kLeaky * e;
  atomicMax(&mx[d], ord_enc(e));
}

__global__ void edge_sum2(const int* __restrict__ ei, const float* __restrict__ es,
                          const float* __restrict__ ed, const unsigned* __restrict__ mx,
                          float* __restrict__ sum) {
  int i = blockIdx.x * blockDim.x + threadIdx.x;
  if (i >= kEtot) return;
  int s, d; edge_sd(ei, i, s, d);
  float e = es[s] + ed[d];
  e = e > 0.f ? e : kLeaky * e;
  atomicAdd(&sum[d], expf(e - ord_dec(mx[d])));
}

__global__ void edge_agg2(const int* __restrict__ ei, const float* __restrict__ es,
                          const float* __restrict__ ed, const unsigned* __restrict__ mx,
                          const float* __restrict__ h2, float* __restrict__ agg) {
  int idx = blockIdx.x * blockDim.x + threadIdx.x;   // edge*kOut + c
  if (idx >= kEtot * kOut) return;
  int i = idx / kOut, c = idx - i * kOut;
  int s, d; edge_sd(ei, i, s, d);
  float e = es[s] + ed[d];
  e = e > 0.f ? e : kLeaky * e;
  float ex = expf(e - ord_dec(mx[d]));
  atomicAdd(&agg[d * kOut + c], h2[s * kOut + c] * ex);
}

// normalize + bias + log_softmax -> d_out
__global__ void finalize2(const float* __restrict__ agg, const float* __restrict__ sum,
                          const float* __restrict__ b2, float* __restrict__ out) {
  int n = blockIdx.x * blockDim.x + threadIdx.x;
  if (n >= kN) return;
  float v[kOut];
  float sinv = 1.f / sum[n];
  float m = -INFINITY;
#pragma unroll
  for (int c = 0; c < kOut; ++c) {
    float t = agg[n * kOut + c] * sinv + b2[c];
    v[c] = t;
    m = fmaxf(m, t);
  }
  float se = 0.f;
#pragma unroll
  for (int c = 0; c < kOut; ++c) se += expf(v[c] - m);
  float lse = m + logf(se);
#pragma unroll
  for (int c = 0; c < kOut; ++c) out[n * kOut + c] = v[c] - lse;
}

extern "C" void kernel_launch(void* const* d_in, const int* in_sizes, int n_in,
                              void* d_out, int out_size, void* d_ws, size_t ws_size,
                              hipStream_t stream) {
  (void)in_sizes; (void)n_in; (void)out_size; (void)ws_size;
  const float* x       = (const float*)d_in[0];
  const int*   ei      = (const int*)  d_in[1];   // [2,E]
  const float* W1      = (const float*)d_in[2];
  const float* a1_src  = (const float*)d_in[3];
  const float* a1_dst  = (const float*)d_in[4];
  const float* b1      = (const float*)d_in[5];
  const float* bn_g    = (const float*)d_in[6];
  const float* bn_b    = (const float*)d_in[7];
  const float* bn_mu   = (const float*)d_in[8];
  const float* bn_var  = (const float*)d_in[9];
  const float* W2      = (const float*)d_in[10];
  const float* a2_src  = (const float*)d_in[11];
  const float* a2_dst  = (const float*)d_in[12];
  const float* b2      = (const float*)d_in[13];
  float* out = (float*)d_out;

  // workspace carve-up (all f32-sized slots)
  float* ws   = (float*)d_ws;
  float* h1   = ws;                         // N*128  (later reused as BN/ELU output)
  float* agg1 = h1   + kN * kD1;            // N*128
  float* es1  = agg1 + kN * kD1;            // N*4
  float* ed1  = es1  + kN * kHeads;         // N*4
  float* sum1 = ed1  + kN * kHeads;         // N*4
  unsigned* mx1 = (unsigned*)(sum1 + kN * kHeads);   // N*4
  float* h2   = (float*)(mx1 + kN * kHeads);// N*40
  float* agg2 = h2   + kN * kOut;           // N*40
  float* es2  = agg2 + kN * kOut;           // N
  float* ed2  = es2  + kN;                  // N
  float* sum2 = ed2  + kN;                  // N
  unsigned* mx2 = (unsigned*)(sum2 + kN);   // N
  float* w1t  = (float*)(mx2 + kN);         // 128*128  (col-major W1)
  float* w2t  = w1t + kD1 * kInCh;          // 48*128   (col-major, zero-padded W2)

  const int B = 256;
  // init accumulators + weight transposes
  init_ws_kernel<<<(kN * kD1 + B - 1) / B, B, 0, stream>>>(agg1, agg2, sum1,
                                                           sum2, mx1, mx2);
  transpose_w1<<<(kD1 * kInCh + B - 1) / B, B, 0, stream>>>(W1, w1t);
  transpose_w2<<<(kOutPad * kD1 + B - 1) / B, B, 0, stream>>>(W2, w2t);
  // layer 1
  gemm1_wmma<<<kN / 16, 128, 0, stream>>>(x, w1t, h1);
  scores1_kernel<<<(kN * kHeads + B - 1) / B, B, 0, stream>>>(h1, a1_src, a1_dst,
                                                              es1, ed1);
  edge_max1<<<(kEtot + B - 1) / B, B, 0, stream>>>(ei, es1, ed1, mx1);
  edge_sum1<<<(kEtot + B - 1) / B, B, 0, stream>>>(ei, es1, ed1, mx1, sum1);
  edge_agg1<<<(kEtot + 7) / 8, B, 0, stream>>>(ei, es1, ed1, mx1, h1, agg1);
  finalize1<<<(kN * kD1 + B - 1) / B, B, 0, stream>>>(agg1, sum1, b1, bn_g, bn_b,
                                                      bn_mu, bn_var, h1);
  // layer 2
  gemm2_wmma<<<kN / 16, 32, 0, stream>>>(h1, w2t, h2);
  scores2_kernel<<<(kN + B - 1) / B, B, 0, stream>>>(h2, a2_src, a2_dst, es2, ed2);
  edge_max2<<<(kEtot + B - 1) / B, B, 0, stream>>>(ei, es2, ed2, mx2);
  edge_sum2<<<(kEtot + B - 1) / B, B, 0, stream>>>(ei, es2, ed2, mx2, sum2);
  edge_agg2<<<(kEtot * kOut + B - 1) / B, B, 0, stream>>>(ei, es2, ed2, mx2, h2,
                                                          agg2);
  finalize2<<<(kN + B - 1) / B, B, 0, stream>>>(agg2, sum2, b2, out);
}